// SoftClusteringModule_4569845203472
// MI455X (gfx1250) — compile-verified
//
#include <hip/hip_runtime.h>
#include <hip/hip_bf16.h>
#include <math.h>

typedef __attribute__((ext_vector_type(2))) float v2f;
typedef __attribute__((ext_vector_type(8))) float v8f;

#define T_TOK   500000
#define NTILES  (T_TOK / 16)   // 31250, exact
#define KCL     8
#define EMB     64
#define PITCH   68             // LDS row pitch in floats (16B-aligned, bank-rotating)

__device__ __forceinline__ v8f wmma_f32(v2f a, v2f b, v8f c) {
  // D(16x16,f32) = A(16x4,f32) * B(4x16,f32) + C
  return __builtin_amdgcn_wmma_f32_16x16x4_f32(false, a, false, b, (short)0, c,
                                               false, false);
}

// ---------------- pass 0: zero workspace accumulators ----------------
__global__ void zero_ws_kernel(double* dacc, float* fzero, int nf) {
  int i = blockIdx.x * blockDim.x + threadIdx.x;
  if (i < 128) dacc[i] = 0.0;
  if (i < nf)  fzero[i] = 0.0f;
}

// ---------------- pass 1: per-dim sum / sumsq (f64) ----------------
__global__ __launch_bounds__(256) void var_kernel(const float* __restrict__ x,
                                                  double* __restrict__ acc) {
  __shared__ double shS[256];
  __shared__ double shQ[256];
  const int tid = threadIdx.x;
  const int d   = tid & 63;        // dim
  const int rg  = tid >> 6;        // row group 0..3
  double s = 0.0, q = 0.0;
  for (int t = blockIdx.x * 4 + rg; t < T_TOK; t += gridDim.x * 4) {
    float v = x[(size_t)t * EMB + d];   // 256 threads read 4 full rows: coalesced
    s += (double)v;
    q += (double)v * (double)v;
  }
  shS[tid] = s; shQ[tid] = q;
  __syncthreads();
  if (tid < 64) {
    double S = shS[tid] + shS[tid + 64] + shS[tid + 128] + shS[tid + 192];
    double Q = shQ[tid] + shQ[tid + 64] + shQ[tid + 128] + shQ[tid + 192];
    atomicAdd(&acc[tid],      S);
    atomicAdd(&acc[64 + tid], Q);
  }
}

// ---------------- pass 2: cluster_count -> mask ----------------
__global__ void mask_kernel(const double* __restrict__ acc, float* __restrict__ maskv) {
  if (threadIdx.x == 0) {
    double tot = 0.0;
    for (int d = 0; d < 64; ++d) {
      double s = acc[d], q = acc[64 + d];
      double var = (q - s * s / (double)T_TOK) / (double)(T_TOK - 1); // ddof=1
      tot += var;
    }
    float vm = (float)(tot / 64.0);
    int cc = (int)(vm * 5.0f + 2.0f);       // trunc == floor (vm >= 0)
    if (cc < 2) cc = 2;
    if (cc > KCL) cc = KCL;
    for (int n = 0; n < KCL; ++n) maskv[n] = (n < cc) ? 1.0f : 0.0f;
  }
}

// ---------------- pass 3: fused dists + gumbel-softmax + both GEMMs ----------------
__global__ __launch_bounds__(256) void cluster_main(
    const float* __restrict__ x, const float* __restrict__ gu,
    const float* __restrict__ cen, const float* __restrict__ maskv,
    float* __restrict__ gsums, float* __restrict__ gmass) {
  __shared__ float tileSh[8 * 16 * PITCH];   // per-wave 16x64 token tile
  __shared__ float xsqSh[8 * 16];            // per-wave token norms
  __shared__ float blkSums[KCL * EMB];
  __shared__ float blkMass[KCL];

  const int tid  = threadIdx.x;
  const int lane = tid & 31;
  const int w    = tid >> 5;                 // wave in block (0..7)
  const int lo   = lane & 15;
  const bool hi  = lane >= 16;
  float* tl = &tileSh[w * 16 * PITCH];

  for (int i = tid; i < KCL * EMB; i += blockDim.x) blkSums[i] = 0.0f;
  if (tid < KCL) blkMass[tid] = 0.0f;
  __syncthreads();

  // Per-lane constants: lane's cluster = lo. Pads clusters 8..15 with zeros.
  float mymask = 0.0f, c2 = 0.0f;
  v2f cb[16];                                // B-matrix (4x16 centers^T) per K-step
  if (lo < KCL) {
    mymask = maskv[lo];
    const float* cr = cen + lo * EMB;
    float s = 0.0f;
    for (int e = 0; e < EMB; ++e) { float v = cr[e]; s += v * v; }
    c2 = s;
#pragma unroll
    for (int s4 = 0; s4 < 16; ++s4) {
      int d = 4 * s4 + (hi ? 2 : 0);
      cb[s4].x = cr[d]; cb[s4].y = cr[d + 1];
    }
  } else {
#pragma unroll
    for (int s4 = 0; s4 < 16; ++s4) { cb[s4].x = 0.0f; cb[s4].y = 0.0f; }
  }

  v8f acc2[4];                               // cluster_sums accumulators (16x64 tile)
#pragma unroll
  for (int j = 0; j < 4; ++j) acc2[j] = (v8f){0.f,0.f,0.f,0.f,0.f,0.f,0.f,0.f};
  float massAcc = 0.0f;

  const int waveGlobal = blockIdx.x * (blockDim.x >> 5) + w;
  const int waveCount  = gridDim.x * (blockDim.x >> 5);

  for (int tile = waveGlobal; tile < NTILES; tile += waveCount) {
    const int t0 = tile * 16;
    {
      int nt = tile + waveCount;
      if (nt < NTILES) __builtin_prefetch(x + (size_t)nt * EMB * 16, 0, 0);
    }
    // ---- stage 16x64 tile to LDS, coalesced b128 ----
    const float4* xv = reinterpret_cast<const float4*>(x + (size_t)t0 * EMB);
#pragma unroll
    for (int it = 0; it < 8; ++it) {
      int idx = it * 32 + lane;
      float4 f4 = xv[idx];
      int tok = idx >> 4, dg = idx & 15;
      *reinterpret_cast<float4*>(&tl[tok * PITCH + dg * 4]) = f4;
    }
    asm volatile("s_wait_dscnt 0" ::: "memory");   // wave-private LDS RAW fence

    // ---- token norms (lanes 0-15, one token each) ----
    if (!hi) {
      float s = 0.0f;
      const float* row = &tl[lo * PITCH];
#pragma unroll
      for (int e = 0; e < EMB; ++e) { float v = row[e]; s += v * v; }
      xsqSh[w * 16 + lo] = s;
    }
    asm volatile("s_wait_dscnt 0" ::: "memory");

    // ---- matmul 1: D[token][cluster] = X(16x64) * C^T(64x16) via 16 WMMAs ----
    v8f dacc = (v8f){0.f,0.f,0.f,0.f,0.f,0.f,0.f,0.f};
#pragma unroll
    for (int s4 = 0; s4 < 16; ++s4) {
      int d = 4 * s4 + (hi ? 2 : 0);
      v2f a; a.x = tl[lo * PITCH + d]; a.y = tl[lo * PITCH + d + 1];
      dacc = wmma_f32(a, cb[s4], dacc);
    }

    // ---- logits + gumbel + 16-lane softmax; rows v = tokens ----
    float asg[8];                             // assign[token mbase+v][cluster lo]
    const int mbase = hi ? 8 : 0;
#pragma unroll
    for (int v = 0; v < 8; ++v) {
      int m = v + mbase;
      float x2 = xsqSh[w * 16 + m];
      float lg = -1e30f;
      if (lo < KCL && mymask > 0.5f) {
        float u  = gu[(size_t)(t0 + m) * KCL + lo];
        float gn = -__logf(-__logf(u));
        float dist = x2 - 2.0f * dacc[v] + c2;
        lg = -dist + gn;                      // CLUSTER_TEMP+1e-8 == 1.0f in fp32
      }
      float mx = lg;
#pragma unroll
      for (int off = 8; off >= 1; off >>= 1) mx = fmaxf(mx, __shfl_xor(mx, off));
      float p = __expf(lg - mx);
      float sm = p;
#pragma unroll
      for (int off = 8; off >= 1; off >>= 1) sm += __shfl_xor(sm, off);
      float a = p / sm;
      asg[v] = a;
      massAcc += a;                           // lane-local: cluster lo mass
    }

    // ---- matmul 2: cluster_sums(16x64) += assign^T(16x16) * X(16x64) ----
#pragma unroll
    for (int s2 = 0; s2 < 4; ++s2) {
      const int r0 = (s2 & 1) * 4;
      v2f a2;
      if (s2 < 2) {      // tokens 0..7: low half owns K, high half fetches via ^16
        float sx = __shfl(asg[r0 + 2], lane ^ 16);
        float sy = __shfl(asg[r0 + 3], lane ^ 16);
        a2.x = hi ? sx : asg[r0 + 0];
        a2.y = hi ? sy : asg[r0 + 1];
      } else {           // tokens 8..15: high half owns K, low half fetches via ^16
        float sx = __shfl(asg[r0 + 0], lane ^ 16);
        float sy = __shfl(asg[r0 + 1], lane ^ 16);
        a2.x = hi ? asg[r0 + 2] : sx;
        a2.y = hi ? asg[r0 + 3] : sy;
      }
      const int t = 4 * s2 + (hi ? 2 : 0);
#pragma unroll
      for (int j = 0; j < 4; ++j) {
        v2f b;
        b.x = tl[t * PITCH + j * 16 + lo];
        b.y = tl[(t + 1) * PITCH + j * 16 + lo];
        acc2[j] = wmma_f32(a2, b, acc2[j]);
      }
    }
  }

  // ---- hierarchical reduction: regs -> block LDS -> global atomics ----
  if (lo < KCL) atomicAdd(&blkMass[lo], massAcc);      // both halves: disjoint tokens
  if (!hi) {                                           // D2 rows 0..7 = clusters 0..7
#pragma unroll
    for (int v = 0; v < KCL; ++v)
#pragma unroll
      for (int j = 0; j < 4; ++j)
        atomicAdd(&blkSums[v * EMB + j * 16 + lo], acc2[j][v]);
  }
  __syncthreads();
  for (int i = tid; i < KCL * EMB; i += blockDim.x) atomicAdd(&gsums[i], blkSums[i]);
  if (tid < KCL) atomicAdd(&gmass[tid], blkMass[tid]);
}

// ---------------- pass 4: finalize (8x64 output) ----------------
__global__ void out_kernel(const float* __restrict__ maskv,
                           const float* __restrict__ gsums,
                           const float* __restrict__ gmass,
                           float* __restrict__ out) {
  int i = blockIdx.x * blockDim.x + threadIdx.x;
  if (i < KCL * EMB) {
    int k = i >> 6;
    out[i] = maskv[k] * (gsums[i] / (gmass[k] + 1e-8f));
  }
}

extern "C" void kernel_launch(void* const* d_in, const int* in_sizes, int n_in,
                              void* d_out, int out_size, void* d_ws, size_t ws_size,
                              hipStream_t stream) {
  const float* x   = (const float*)d_in[0];   // token_embs (T,64)
  const float* gu  = (const float*)d_in[1];   // gumbel_u   (T,8)
  const float* cen = (const float*)d_in[2];   // centers    (8,64)
  float* out = (float*)d_out;                 // (8,64)

  // workspace layout: 128 f64 (sum/sumsq) | 8 f32 mask | 512 f32 sums | 8 f32 mass
  double* dacc  = (double*)d_ws;
  float*  fbase = (float*)(dacc + 128);
  float*  maskv = fbase;
  float*  gsums = fbase + 8;
  float*  gmass = fbase + 8 + 512;

  zero_ws_kernel<<<3, 256, 0, stream>>>(dacc, fbase, 8 + 512 + 8);
  var_kernel<<<512, 256, 0, stream>>>(x, dacc);
  mask_kernel<<<1, 64, 0, stream>>>(dacc, maskv);
  cluster_main<<<256, 256, 0, stream>>>(x, gu, cen, maskv, gsums, gmass);
  out_kernel<<<2, 256, 0, stream>>>(maskv, gsums, gmass, out);
}